// ANet_3977139716220
// MI455X (gfx1250) — compile-verified
//
#include <hip/hip_runtime.h>

// ---------------------------------------------------------------------------
// Fused ANet forward for MI455X (gfx1250, wave32):
//   h = relu(x@W1^T + b1); t = tanh(h@W2^T + b2); z = t@Wopt^T - bopt;
//   y = proj_capped_simplex(z, u, 90)   (output [B,32] fp32)
// GEMMs via V_WMMA_F32_16X16X4_F32 (fp32-exact); projection lane-per-sample
// with v_med3_f32 clamps (clip(d,0,u) == med3(d,0,u) since u>=0).
// ---------------------------------------------------------------------------

#define S_DIM   64
#define A_DIM   32
#define HIDDEN  30
#define BUDGETF 90.0f
#define NBISECT 32     // fp32 bracket (width ~10) is ulp-tight after ~30 halvings
#define NNEWTON 3      // semismooth-Newton polish -> exact fp32 lambda

#define WAVES          4
#define TPB            (WAVES * 32)      // 128 threads = 4 wave32 waves
#define ROWS_PER_WAVE  32
#define ROWS_PER_BLOCK (WAVES * ROWS_PER_WAVE)  // 128 samples per block

typedef __attribute__((ext_vector_type(2))) float v2f;
typedef __attribute__((ext_vector_type(8))) float v8f;

static __device__ __forceinline__ v8f wmma4(v2f a, v2f b, v8f c) {
    // D = A(16x4) * B(4x16) + C(16x16), fp32 throughout.
    return __builtin_amdgcn_wmma_f32_16x16x4_f32(
        /*neg_a=*/false, a, /*neg_b=*/false, b,
        /*c_mod=*/(short)0, c, /*reuse_a=*/false, /*reuse_b=*/false);
}

// clip(d, 0, u) for u >= 0 : single v_med3_f32
static __device__ __forceinline__ float clip0u(float d, float u) {
    return __builtin_amdgcn_fmed3f(d, 0.0f, u);
}

__launch_bounds__(TPB)
__global__ void anet_fused_kernel(const float* __restrict__ x,
                                  const float* __restrict__ W1,
                                  const float* __restrict__ b1,
                                  const float* __restrict__ W2,
                                  const float* __restrict__ b2,
                                  const float* __restrict__ Wopt,
                                  const float* __restrict__ bopt,
                                  const float* __restrict__ u,
                                  float* __restrict__ out,
                                  int Btot)
{
    // Zero-padded weight tiles (so WMMA B-fragments need no guards).
    __shared__ __align__(16) float W1p[32][64];   // [n][k], rows 30..31 = 0
    __shared__ __align__(16) float W2p[32][32];   // [n][k], cols 30..31 = 0
    __shared__ __align__(16) float Wop[32][32];   // [n][k]
    __shared__ __align__(16) float b1p[32];
    __shared__ __align__(16) float b2p[32];
    __shared__ __align__(16) float bop[32];
    __shared__ __align__(16) float up [32];
    // Per-wave staging tiles. Strides (34, 33 floats) picked for LDS bank spread:
    //  - hbuf b64 fragment reads: row*34 -> even offsets, rows land on distinct banks
    //  - zbuf lane-per-sample reads: gcd(33,64)=1 -> 32 distinct banks, conflict-free
    __shared__ __align__(16) float hbuf[WAVES][16][34];
    __shared__ __align__(16) float zbuf[WAVES][32][33];

    const int tid = threadIdx.x;

    for (int i = tid; i < 32 * 64; i += TPB) {
        int n = i >> 6, k = i & 63;
        W1p[n][k] = (n < HIDDEN) ? W1[n * S_DIM + k] : 0.0f;
    }
    for (int i = tid; i < 32 * 32; i += TPB) {
        int n = i >> 5, k = i & 31;
        W2p[n][k] = (k < HIDDEN) ? W2[n * HIDDEN + k] : 0.0f;
        Wop[n][k] = Wopt[n * A_DIM + k];
    }
    if (tid < 32) {
        b1p[tid] = (tid < HIDDEN) ? b1[tid] : 0.0f;
        b2p[tid] = b2[tid];
        bop[tid] = bopt[tid];
        up [tid] = u[tid];
    }
    __syncthreads();

    const int w  = tid >> 5;        // wave id in block
    const int l  = tid & 31;        // lane
    const int lr = l & 15;          // lane % 16
    const int lh = l >> 4;          // lane half (0/1)

    const int waveRow0 = blockIdx.x * ROWS_PER_BLOCK + w * ROWS_PER_WAVE;

    for (int sub = 0; sub < 2; ++sub) {
        const int r0 = waveRow0 + sub * 16;
        int arow = r0 + lr;
        if (arow >= Btot) arow = Btot - 1;           // clamp (tail safety)
        const float* xrow = x + (size_t)arow * S_DIM + 2 * lh;
        __builtin_prefetch(xrow, 0, 1);              // global_prefetch_b8

        // ---- stage 1: h = relu(x @ W1^T + b1)   M=16, K=64, N=32(pad) ----
        v8f acc0 = {0.f,0.f,0.f,0.f,0.f,0.f,0.f,0.f};
        v8f acc1 = {0.f,0.f,0.f,0.f,0.f,0.f,0.f,0.f};
        #pragma unroll
        for (int k = 0; k < 16; ++k) {
            v2f a  = *(const v2f*)(xrow + 4 * k);                    // A: M=lr, K=4k+2lh+{0,1}
            v2f bA = *(const v2f*)(&W1p[lr     ][4 * k + 2 * lh]);   // B: N=lr
            v2f bB = *(const v2f*)(&W1p[16 + lr][4 * k + 2 * lh]);   // B: N=16+lr
            acc0 = wmma4(a, bA, acc0);
            acc1 = wmma4(a, bB, acc1);
        }
        {
            const float biasA = b1p[lr], biasB = b1p[16 + lr];
            #pragma unroll
            for (int r = 0; r < 8; ++r) {
                const int m = r + 8 * lh;                            // D row
                hbuf[w][m][lr]      = fmaxf(acc0[r] + biasA, 0.0f);
                hbuf[w][m][16 + lr] = fmaxf(acc1[r] + biasB, 0.0f);
            }
        }
        __builtin_amdgcn_wave_barrier();

        // ---- stage 2: t = tanh(h @ W2^T + b2)   M=16, K=32(pad), N=32 ----
        acc0 = (v8f){0.f,0.f,0.f,0.f,0.f,0.f,0.f,0.f};
        acc1 = (v8f){0.f,0.f,0.f,0.f,0.f,0.f,0.f,0.f};
        #pragma unroll
        for (int k = 0; k < 8; ++k) {
            v2f a  = *(const v2f*)(&hbuf[w][lr][4 * k + 2 * lh]);
            v2f bA = *(const v2f*)(&W2p[lr     ][4 * k + 2 * lh]);
            v2f bB = *(const v2f*)(&W2p[16 + lr][4 * k + 2 * lh]);
            acc0 = wmma4(a, bA, acc0);
            acc1 = wmma4(a, bB, acc1);
        }
        __builtin_amdgcn_wave_barrier();
        {
            const float biasA = b2p[lr], biasB = b2p[16 + lr];
            #pragma unroll
            for (int r = 0; r < 8; ++r) {
                const int m = r + 8 * lh;
                hbuf[w][m][lr]      = tanhf(acc0[r] + biasA);        // reuse hbuf for t
                hbuf[w][m][16 + lr] = tanhf(acc1[r] + biasB);
            }
        }
        __builtin_amdgcn_wave_barrier();

        // ---- stage 3: z = t @ Wopt^T - bopt   M=16, K=32, N=32 ----
        acc0 = (v8f){0.f,0.f,0.f,0.f,0.f,0.f,0.f,0.f};
        acc1 = (v8f){0.f,0.f,0.f,0.f,0.f,0.f,0.f,0.f};
        #pragma unroll
        for (int k = 0; k < 8; ++k) {
            v2f a  = *(const v2f*)(&hbuf[w][lr][4 * k + 2 * lh]);
            v2f bA = *(const v2f*)(&Wop[lr     ][4 * k + 2 * lh]);
            v2f bB = *(const v2f*)(&Wop[16 + lr][4 * k + 2 * lh]);
            acc0 = wmma4(a, bA, acc0);
            acc1 = wmma4(a, bB, acc1);
        }
        {
            const float biasA = bop[lr], biasB = bop[16 + lr];
            #pragma unroll
            for (int r = 0; r < 8; ++r) {
                const int m = r + 8 * lh;
                zbuf[w][sub * 16 + m][lr]      = acc0[r] - biasA;
                zbuf[w][sub * 16 + m][16 + lr] = acc1[r] - biasB;
            }
        }
        __builtin_amdgcn_wave_barrier();
    }

    // ---- stage 4: capped-simplex projection, one lane per sample ----
    float z[A_DIM], uu[A_DIM];
    #pragma unroll
    for (int d = 0; d < A_DIM; ++d) {
        z[d]  = zbuf[w][l][d];     // conflict-free: stride 33, gcd(33,64)=1
        uu[d] = up[d];
    }

    float lo = z[0] - uu[0], hi = z[0];
    #pragma unroll
    for (int d = 1; d < A_DIM; ++d) {
        lo = fminf(lo, z[d] - uu[d]);
        hi = fmaxf(hi, z[d]);
    }

    #pragma unroll 1
    for (int it = 0; it < NBISECT; ++it) {
        const float mid = 0.5f * (lo + hi);
        float g = -BUDGETF;
        #pragma unroll
        for (int d = 0; d < A_DIM; ++d)
            g += clip0u(z[d] - mid, uu[d]);          // v_med3_f32
        const bool gt = g > 0.0f;
        lo = gt ? mid : lo;
        hi = gt ? hi : mid;
    }
    float lam = 0.5f * (lo + hi);

    #pragma unroll 1
    for (int it = 0; it < NNEWTON; ++it) {
        float g = -BUDGETF, act = 0.0f;
        #pragma unroll
        for (int d = 0; d < A_DIM; ++d) {
            const float dd = z[d] - lam;
            g += clip0u(dd, uu[d]);                  // v_med3_f32
            act += (dd > 0.0f && dd < uu[d]) ? 1.0f : 0.0f;
        }
        const float gp = (act == 0.0f) ? -1.0f : -act;
        lam = lam - g / gp;
    }

    const int gr = waveRow0 + l;
    if (gr < Btot) {
        float* orow = out + (size_t)gr * A_DIM;
        #pragma unroll
        for (int j = 0; j < A_DIM / 4; ++j) {
            float4 y4;
            y4.x = clip0u(z[4 * j + 0] - lam, uu[4 * j + 0]);
            y4.y = clip0u(z[4 * j + 1] - lam, uu[4 * j + 1]);
            y4.z = clip0u(z[4 * j + 2] - lam, uu[4 * j + 2]);
            y4.w = clip0u(z[4 * j + 3] - lam, uu[4 * j + 3]);
            *(float4*)(orow + 4 * j) = y4;
        }
    }
}

extern "C" void kernel_launch(void* const* d_in, const int* in_sizes, int n_in,
                              void* d_out, int out_size, void* d_ws, size_t ws_size,
                              hipStream_t stream) {
    const float* x    = (const float*)d_in[0];
    const float* W1   = (const float*)d_in[1];
    const float* b1   = (const float*)d_in[2];
    const float* W2   = (const float*)d_in[3];
    const float* b2   = (const float*)d_in[4];
    const float* Wopt = (const float*)d_in[5];
    const float* bopt = (const float*)d_in[6];
    const float* u    = (const float*)d_in[7];
    float* out = (float*)d_out;

    const int Btot = in_sizes[0] / S_DIM;
    const int grid = (Btot + ROWS_PER_BLOCK - 1) / ROWS_PER_BLOCK;
    anet_fused_kernel<<<grid, TPB, 0, stream>>>(x, W1, b1, W2, b2, Wopt, bopt, u,
                                                out, Btot);
}